// TorchSTFT_31473520345493
// MI455X (gfx1250) — compile-verified
//
#include <hip/hip_runtime.h>
#include <math.h>

#define FILT   800
#define HOP    200
#define WIN    800
#define BATCH  16
#define T_LEN  160000
#define PAD    400              // FILT/2
#define NFRM   801              // (T_LEN + 2*PAD - WIN)/HOP + 1
#define M_ROWS (BATCH * NFRM)   // 12816 valid GEMM rows
#define MBLK   4                // M-tiles per wave (register blocking)
#define M_TILES_PAD 804         // 801 padded up to multiple of MBLK
#define M_PAD  (M_TILES_PAD * 16)   // 12864 rows allocated
#define K_DIM  800
#define N_DIM  800
#define TWO_PI 6.28318530717958647692f

typedef __attribute__((ext_vector_type(2))) float v2f;
typedef __attribute__((ext_vector_type(8))) float v8f;

// ---------------------------------------------------------------------------
// DFT twiddle tables: Tc[n*800+k] = cos(2*pi*n*k/800), Ts = sin(...).
// Integer modulo keeps the trig argument in [0, 2*pi) for full accuracy.
// ---------------------------------------------------------------------------
__global__ void build_tables_kernel(float* __restrict__ Tc, float* __restrict__ Ts) {
    int idx = blockIdx.x * blockDim.x + threadIdx.x;
    if (idx >= FILT * FILT) return;
    int n = idx / FILT;
    int k = idx - n * FILT;
    int m = (n * k) % FILT;
    float ang = TWO_PI * (float)m / (float)FILT;
    Tc[idx] = cosf(ang);
    Ts[idx] = sinf(ang);
}

// ---------------------------------------------------------------------------
// Frame extraction with reflect padding + Hann analysis window.
// wf is M_PAD x 800 row-major (only first M_ROWS rows written), row = b*NFRM+f.
// ---------------------------------------------------------------------------
__global__ void frame_window_kernel(const float* __restrict__ x, float* __restrict__ wf) {
    int idx = blockIdx.x * blockDim.x + threadIdx.x;
    if (idx >= M_ROWS * WIN) return;
    int row = idx / WIN;
    int n   = idx - row * WIN;
    int b   = row / NFRM;
    int f   = row - b * NFRM;
    int q = f * HOP + n - PAD;                       // position in unpadded signal
    if (q < 0)            q = -q;                    // reflect (no edge repeat)
    else if (q >= T_LEN)  q = 2 * T_LEN - 2 - q;
    float w = 0.5f - 0.5f * cosf(TWO_PI * (float)n / (float)WIN);
    wf[idx] = x[b * T_LEN + q] * w;
}

// ---------------------------------------------------------------------------
// Forward DFT GEMM: FR = wf @ Tc ; G = wf @ Ts   (reference full_i == -G).
// Each wave: 4 M-tiles x 1 N-tile, 8 f32 accumulators. B fragments (Tc/Ts)
// are reused across the 4 M-tiles -> ~1 VMEM per WMMA in the K loop.
// ---------------------------------------------------------------------------
__global__ __launch_bounds__(128)
void gemm_fwd_kernel(const float* __restrict__ A,
                     const float* __restrict__ Tc,
                     const float* __restrict__ Ts,
                     float* __restrict__ FR,
                     float* __restrict__ G) {
    const int lane   = threadIdx.x & 31;
    const int wave   = threadIdx.x >> 5;
    const int tileM0 = blockIdx.x * MBLK;            // 0..800 step 4 (padded)
    const int tileN  = blockIdx.y * 4 + wave;        // 0..49 valid
    if (tileN >= N_DIM / 16) return;

    const int half = lane >> 4;                      // 0: K pair {0,1}, 1: {2,3}
    const int l15  = lane & 15;
    const int col  = tileN * 16 + l15;

    const float* __restrict__ Abase = A + (size_t)(tileM0 * 16 + l15) * K_DIM + 2 * half;
    const float* __restrict__ TcB   = Tc + (size_t)(2 * half) * N_DIM + col;
    const float* __restrict__ TsB   = Ts + (size_t)(2 * half) * N_DIM + col;

    v8f accR[MBLK], accI[MBLK];
#pragma unroll
    for (int i = 0; i < MBLK; ++i) { accR[i] = (v8f){}; accI[i] = (v8f){}; }

    for (int k = 0; k < K_DIM; k += 4) {
        v2f bc; bc[0] = TcB[(size_t)k * N_DIM];
                bc[1] = TcB[(size_t)(k + 1) * N_DIM];
        v2f bs; bs[0] = TsB[(size_t)k * N_DIM];
                bs[1] = TsB[(size_t)(k + 1) * N_DIM];
        v2f a[MBLK];
#pragma unroll
        for (int i = 0; i < MBLK; ++i) {
            a[i][0] = Abase[(size_t)i * 16 * K_DIM + k];
            a[i][1] = Abase[(size_t)i * 16 * K_DIM + k + 1];
        }
#pragma unroll
        for (int i = 0; i < MBLK; ++i) {
            accR[i] = __builtin_amdgcn_wmma_f32_16x16x4_f32(false, a[i], false, bc,
                                                            (short)0, accR[i], false, false);
            accI[i] = __builtin_amdgcn_wmma_f32_16x16x4_f32(false, a[i], false, bs,
                                                            (short)0, accI[i], false, false);
        }
    }
#pragma unroll
    for (int i = 0; i < MBLK; ++i) {
#pragma unroll
        for (int r = 0; r < 8; ++r) {
            const int row = (tileM0 + i) * 16 + r + 8 * half;   // C/D layout (ISA 7.12.2)
            FR[(size_t)row * N_DIM + col] = accR[i][r];
            G [(size_t)row * N_DIM + col] = accI[i][r];
        }
    }
}

// ---------------------------------------------------------------------------
// Inverse DFT GEMM fused with 1/FILT and the synthesis window:
//   TF = ((FR @ Tc) + (G @ Ts)) * hann(col) * sqrt(0.75+1e-8) / 800
// Same 4x M-tile register blocking; two WMMAs per tile per K-step into one
// accumulator each.
// ---------------------------------------------------------------------------
__global__ __launch_bounds__(128)
void gemm_inv_kernel(const float* __restrict__ FR,
                     const float* __restrict__ G,
                     const float* __restrict__ Tc,
                     const float* __restrict__ Ts,
                     float* __restrict__ TF) {
    const int lane   = threadIdx.x & 31;
    const int wave   = threadIdx.x >> 5;
    const int tileM0 = blockIdx.x * MBLK;
    const int tileN  = blockIdx.y * 4 + wave;
    if (tileN >= N_DIM / 16) return;

    const int half = lane >> 4;
    const int l15  = lane & 15;
    const int col  = tileN * 16 + l15;

    const float* __restrict__ FRbase = FR + (size_t)(tileM0 * 16 + l15) * K_DIM + 2 * half;
    const float* __restrict__ Gbase  = G  + (size_t)(tileM0 * 16 + l15) * K_DIM + 2 * half;
    const float* __restrict__ TcB    = Tc + (size_t)(2 * half) * N_DIM + col;
    const float* __restrict__ TsB    = Ts + (size_t)(2 * half) * N_DIM + col;

    v8f acc[MBLK];
#pragma unroll
    for (int i = 0; i < MBLK; ++i) acc[i] = (v8f){};

    for (int k = 0; k < K_DIM; k += 4) {
        v2f bc; bc[0] = TcB[(size_t)k * N_DIM];
                bc[1] = TcB[(size_t)(k + 1) * N_DIM];
        v2f bs; bs[0] = TsB[(size_t)k * N_DIM];
                bs[1] = TsB[(size_t)(k + 1) * N_DIM];
#pragma unroll
        for (int i = 0; i < MBLK; ++i) {
            v2f ar; ar[0] = FRbase[(size_t)i * 16 * K_DIM + k];
                    ar[1] = FRbase[(size_t)i * 16 * K_DIM + k + 1];
            v2f ag; ag[0] = Gbase[(size_t)i * 16 * K_DIM + k];
                    ag[1] = Gbase[(size_t)i * 16 * K_DIM + k + 1];
            acc[i] = __builtin_amdgcn_wmma_f32_16x16x4_f32(false, ar, false, bc,
                                                           (short)0, acc[i], false, false);
            acc[i] = __builtin_amdgcn_wmma_f32_16x16x4_f32(false, ag, false, bs,
                                                           (short)0, acc[i], false, false);
        }
    }
    // sw[n] = hann(n)/norm with norm = sqrt(1/(1-HOP/WIN+1e-8)); t scaled 1/FILT
    const float hw    = 0.5f - 0.5f * cosf(TWO_PI * (float)col / (float)WIN);
    const float scale = hw * sqrtf(0.75f + 1e-8f) * (1.0f / (float)FILT);
#pragma unroll
    for (int i = 0; i < MBLK; ++i) {
#pragma unroll
        for (int r = 0; r < 8; ++r) {
            const int row = (tileM0 + i) * 16 + r + 8 * half;
            TF[(size_t)row * N_DIM + col] = acc[i][r] * scale;
        }
    }
}

// ---------------------------------------------------------------------------
// Overlap-add as a gather (each output sample sums <=4 frame contributions),
// fused with the crop [PAD, PAD+T_LEN). No atomics needed.
// ---------------------------------------------------------------------------
__global__ void ola_kernel(const float* __restrict__ TF, float* __restrict__ out) {
    int idx = blockIdx.x * blockDim.x + threadIdx.x;
    if (idx >= BATCH * T_LEN) return;
    int b = idx / T_LEN;
    int s = idx - b * T_LEN;
    int p = s + PAD;                                  // position in padded OLA
    int f_hi = p / HOP;            if (f_hi > NFRM - 1) f_hi = NFRM - 1;
    int num  = p - (WIN - 1);
    int f_lo = (num > 0) ? (num + HOP - 1) / HOP : 0; // ceil(num/HOP), clamped
    float acc = 0.0f;
    for (int f = f_lo; f <= f_hi; ++f) {
        int n = p - f * HOP;                          // 0..WIN-1 by construction
        acc += TF[(size_t)(b * NFRM + f) * WIN + n];
    }
    out[idx] = acc;
}

// ---------------------------------------------------------------------------
// Launcher. Workspace layout (floats), GEMM buffers padded to M_PAD rows:
//   Tc[640000] | Ts[640000] | wf[M_PAD*800] | FR[M_PAD*800] | G[M_PAD*800]
// TF reuses the wf region (wf dead after gemm_fwd). Total ~128.7 MB.
// Padded rows (12816..12863) hold defined-but-garbage data; their GEMM
// outputs are never read by the OLA gather.
// ---------------------------------------------------------------------------
extern "C" void kernel_launch(void* const* d_in, const int* in_sizes, int n_in,
                              void* d_out, int out_size, void* d_ws, size_t ws_size,
                              hipStream_t stream) {
    const float* x   = (const float*)d_in[0];
    float*       out = (float*)d_out;

    float* ws = (float*)d_ws;
    const size_t TBL = (size_t)FILT * FILT;        // 640000
    const size_t MAT = (size_t)M_PAD * K_DIM;      // 10291200
    float* Tc = ws;
    float* Ts = Tc + TBL;
    float* wf = Ts + TBL;
    float* FR = wf + MAT;
    float* G  = FR + MAT;
    float* TF = wf;                                // reuse: wf dead after fwd GEMM

    {   // twiddle tables
        int n = FILT * FILT;
        build_tables_kernel<<<(n + 255) / 256, 256, 0, stream>>>(Tc, Ts);
    }
    {   // framing + analysis window
        int n = M_ROWS * WIN;
        frame_window_kernel<<<(n + 255) / 256, 256, 0, stream>>>(x, wf);
    }
    {   // forward DFT (two fused GEMMs), 4x M-tile register blocking
        dim3 grid(M_TILES_PAD / MBLK, (N_DIM / 16 + 3) / 4);   // (201, 13)
        gemm_fwd_kernel<<<grid, 128, 0, stream>>>(wf, Tc, Ts, FR, G);
    }
    {   // inverse DFT (fused GEMM pair) + window/scale
        dim3 grid(M_TILES_PAD / MBLK, (N_DIM / 16 + 3) / 4);
        gemm_inv_kernel<<<grid, 128, 0, stream>>>(FR, G, Tc, Ts, TF);
    }
    {   // overlap-add + crop
        int n = BATCH * T_LEN;
        ola_kernel<<<(n + 255) / 256, 256, 0, stream>>>(TF, out);
    }
}